// seg_loss_12730283065459
// MI455X (gfx1250) — compile-verified
//
#include <hip/hip_runtime.h>
#include <hip/hip_bf16.h>

typedef __attribute__((ext_vector_type(2))) float v2f;
typedef __attribute__((ext_vector_type(8))) float v8f;

#define NBATCH        32
#define N_PER_BATCH   (1 << 20)                       // 1024*1024 elements per batch
#define BLOCKS_PER_B  256
#define NTHREADS      256
#define NWAVES        (NTHREADS / 32)                 // 8 waves per block
#define F4_PER_BATCH  (N_PER_BATCH / 4)               // 262144 float4 per batch
#define THREADS_PER_B (BLOCKS_PER_B * NTHREADS)       // 65536
#define ITERS         (F4_PER_BATCH / THREADS_PER_B)  // 4 float4 iterations per thread

// Full-wave (32-lane) all-reduce sum using V_WMMA_F32_16X16X4_F32.
// A: per-lane value in A-VGPR0 (K=0 for lanes 0-15, K=2 for lanes 16-31), K=1/3 zero.
// B: all-ones 4x16.  =>  D[m][n] = v_m + v_{m+16} for all n.
// Per-lane sum of the 8 D VGPRs gives the half-sum of rows it holds;
// one shfl_xor(16) combines the two halves. Requires EXEC all ones (all call
// sites run with full blocks / a full wave, no divergence).
__device__ __forceinline__ float wave_allreduce_wmma(float v) {
    v2f a; a[0] = v;    a[1] = 0.0f;
    v2f b; b[0] = 1.0f; b[1] = 1.0f;
    v8f c = {};
    v8f d = __builtin_amdgcn_wmma_f32_16x16x4_f32(
        /*neg_a=*/false, a, /*neg_b=*/false, b,
        /*c_mod=*/(short)0, c, /*reuse_a=*/false, /*reuse_b=*/false);
    float s = ((d[0] + d[1]) + (d[2] + d[3])) + ((d[4] + d[5]) + (d[6] + d[7]));
    s += __shfl_xor(s, 16, 32);
    return s; // valid in every lane
}

__device__ __forceinline__ void acc_elem(float xx, float yy,
                                         float& s1, float& s0, float& so,
                                         float& n1, float& n0) {
    float lx  = __logf(xx);          // x in (1e-4, 1-1e-4): both logs are finite
    float l1x = __logf(1.0f - xx);
    float bce = -(yy * lx + (1.0f - yy) * l1x); // y==1 -> -log x ; y==0 -> -log(1-x)
    bool is1 = (yy == 1.0f);
    bool is0 = (yy == 0.0f);
    s1 += is1 ? bce : 0.0f;
    s0 += is0 ? bce : 0.0f;
    so += (!is1 && !is0) ? bce : 0.0f;  // weight-1 bucket (empty for binary masks)
    n1 += is1 ? 1.0f : 0.0f;
    n0 += is0 ? 1.0f : 0.0f;
}

// Kernel 1: streaming pass. grid = (BLOCKS_PER_B, NBATCH). One batch per grid.y.
// Writes 5 partial sums per block: [s1, s0, s_other, n1, n0].
__global__ void seg_partial_kernel(const float* __restrict__ x,
                                   const float* __restrict__ y,
                                   float* __restrict__ partials) {
    const int b = blockIdx.y;
    const float4* __restrict__ x4 =
        (const float4*)(x + (long long)b * N_PER_BATCH);
    const float4* __restrict__ y4 =
        (const float4*)(y + (long long)b * N_PER_BATCH);

    const int t = blockIdx.x * NTHREADS + threadIdx.x; // 0 .. THREADS_PER_B-1

    float s1 = 0.0f, s0 = 0.0f, so = 0.0f, n1 = 0.0f, n0 = 0.0f;

#pragma unroll
    for (int i = 0; i < ITERS; ++i) {
        const int idx = t + i * THREADS_PER_B;
        if (i + 1 < ITERS) { // hint next stride into cache (global_prefetch_b8)
            __builtin_prefetch(&x4[idx + THREADS_PER_B], 0, 1);
            __builtin_prefetch(&y4[idx + THREADS_PER_B], 0, 1);
        }
        float4 xv = x4[idx];  // global_load_b128
        float4 yv = y4[idx];
        acc_elem(xv.x, yv.x, s1, s0, so, n1, n0);
        acc_elem(xv.y, yv.y, s1, s0, so, n1, n0);
        acc_elem(xv.z, yv.z, s1, s0, so, n1, n0);
        acc_elem(xv.w, yv.w, s1, s0, so, n1, n0);
    }

    // In-wave WMMA all-reduce for each accumulator, then LDS combine across 8 waves.
    float r[5] = { s1, s0, so, n1, n0 };
#pragma unroll
    for (int j = 0; j < 5; ++j) r[j] = wave_allreduce_wmma(r[j]);

    __shared__ float smem[NWAVES][5];
    const int wave = threadIdx.x >> 5;
    const int lane = threadIdx.x & 31;
    if (lane == 0) {
#pragma unroll
        for (int j = 0; j < 5; ++j) smem[wave][j] = r[j];
    }
    __syncthreads();

    if (threadIdx.x < 5) {
        float tot = 0.0f;
#pragma unroll
        for (int w = 0; w < NWAVES; ++w) tot += smem[w][threadIdx.x];
        partials[((long long)b * BLOCKS_PER_B + blockIdx.x) * 5 + threadIdx.x] = tot;
    }
}

// Kernel 2: fold the 256 block-partials of one batch, apply class balancing.
// grid = NBATCH blocks, 256 threads (exactly one partial row per thread).
__global__ void seg_combine_kernel(const float* __restrict__ partials,
                                   float* __restrict__ batch_loss) {
    const int b = blockIdx.x;
    const float* p = partials + ((long long)b * BLOCKS_PER_B + threadIdx.x) * 5;

    float r[5];
#pragma unroll
    for (int j = 0; j < 5; ++j) r[j] = p[j];
#pragma unroll
    for (int j = 0; j < 5; ++j) r[j] = wave_allreduce_wmma(r[j]);

    __shared__ float smem[NWAVES][5];
    const int wave = threadIdx.x >> 5;
    const int lane = threadIdx.x & 31;
    if (lane == 0) {
#pragma unroll
        for (int j = 0; j < 5; ++j) smem[wave][j] = r[j];
    }
    __syncthreads();

    if (threadIdx.x == 0) {
        float tot[5] = { 0.0f, 0.0f, 0.0f, 0.0f, 0.0f };
#pragma unroll
        for (int w = 0; w < NWAVES; ++w)
#pragma unroll
            for (int j = 0; j < 5; ++j) tot[j] += smem[w][j];
        const float s1 = tot[0], s0 = tot[1], so = tot[2];
        const float n1 = tot[3], n0 = tot[4];
        batch_loss[b] = s1 / n1 + s0 / n0 + so; // LOSS_WEIGHT == 1.0
    }
}

// Kernel 3: one full wave sums the 32 per-batch losses.
__global__ void seg_final_kernel(const float* __restrict__ batch_loss,
                                 float* __restrict__ out) {
    float v = batch_loss[threadIdx.x];      // 32 threads == one wave32
    float s = wave_allreduce_wmma(v);
    if (threadIdx.x == 0) out[0] = s;
}

extern "C" void kernel_launch(void* const* d_in, const int* in_sizes, int n_in,
                              void* d_out, int out_size, void* d_ws, size_t ws_size,
                              hipStream_t stream) {
    (void)in_sizes; (void)n_in; (void)out_size; (void)ws_size;
    const float* x = (const float*)d_in[0];
    const float* y = (const float*)d_in[1];
    float* out = (float*)d_out;

    float* partials   = (float*)d_ws;                              // 32*256*5 floats
    float* batch_loss = partials + (size_t)NBATCH * BLOCKS_PER_B * 5; // +32 floats

    dim3 grid1(BLOCKS_PER_B, NBATCH);
    seg_partial_kernel<<<grid1, NTHREADS, 0, stream>>>(x, y, partials);
    seg_combine_kernel<<<NBATCH, NTHREADS, 0, stream>>>(partials, batch_loss);
    seg_final_kernel<<<1, 32, 0, stream>>>(batch_loss, out);
}